// GraphVAE_5420248728013
// MI455X (gfx1250) — compile-verified
//
#include <hip/hip_runtime.h>
#include <hip/hip_bf16.h>
#include <math.h>

// ---------------- problem constants (from reference) ----------------
#define BN    64          // batch
#define NN    64          // nodes
#define FD    16          // feature dim
#define HH    512         // hidden
#define LL    128         // latent
#define NE    2016        // N*(N-1)/2
#define ROWS  (BN * NN)   // 4096

typedef __attribute__((ext_vector_type(2))) float v2f;
typedef __attribute__((ext_vector_type(8))) float v8f;

// =====================================================================
// Generic batched GEMM:  C[b] = act(A[b] (MxK) @ B[b] (KxN) + bias)
// fp32 WMMA 16x16x4; each wave32 computes a 16 x (16*NT) C strip.
// Software-pipelined: while the NT WMMAs of K-step k issue, the A/B
// fragments of step k+1 are already in flight in a second register set.
//   A 16x4 tile : 2 VGPR (lanes 0-15: K=k0,k0+1 ; lanes 16-31: K=k0+2,k0+3)
//   B 4x16 tile : 2 VGPR (VGPR0 rows k0 / k0+2, VGPR1 rows k0+1 / k0+3)
//   C 16x16     : 8 VGPR (VGPR i -> rows i / i+8, col = lane&15)
// act: 0=none 1=relu 2=sigmoid
// =====================================================================
template <int NT>
__global__ __launch_bounds__(256) void wmma_gemm_kernel(
    const float* __restrict__ A, const float* __restrict__ Bm,
    const float* __restrict__ bias, float* __restrict__ C,
    int M, int Nn, int K, int sA, int sB, int sC, int act)
{
    int batch = blockIdx.y;
    A  += (size_t)batch * sA;
    Bm += (size_t)batch * sB;
    C  += (size_t)batch * sC;

    int wave = threadIdx.x >> 5;            // 8 waves / block
    int lane = threadIdx.x & 31;
    int tilesM  = M >> 4;
    int stripsN = (Nn >> 4) / NT;
    int strip = blockIdx.x * 8 + wave;      // wave-uniform -> EXEC all 1s
    if (strip >= tilesM * stripsN) return;

    int tm  = (strip / stripsN) << 4;
    int tn  = (strip % stripsN) * (16 * NT);
    int r16 = lane & 15;
    int hi  = lane >> 4;

    v8f acc[NT];
#pragma unroll
    for (int q = 0; q < NT; ++q) acc[q] = (v8f){};

    const float* Arow  = A  + (size_t)(tm + r16) * K;
    const float* Bbase = Bm + tn + r16;

    // ---- prologue: load fragments for k0 = 0 ----
    v2f av, bv[NT];
    {
        int ka = hi << 1;
        av.x = Arow[ka];
        av.y = Arow[ka + 1];
        const float* Bp = Bbase + (size_t)ka * Nn;
#pragma unroll
        for (int q = 0; q < NT; ++q) {
            bv[q].x = Bp[q * 16];
            bv[q].y = Bp[q * 16 + Nn];
        }
    }

#pragma unroll 2
    for (int k0 = 0; k0 < K; k0 += 4) {
        // ---- prefetch next K-step into a second register set ----
        v2f avn, bvn[NT];
        int kn = k0 + 4;
        if (kn < K) {
            int ka = kn + (hi << 1);
            avn.x = Arow[ka];
            avn.y = Arow[ka + 1];
            const float* Bp = Bbase + (size_t)ka * Nn;
#pragma unroll
            for (int q = 0; q < NT; ++q) {
                bvn[q].x = Bp[q * 16];
                bvn[q].y = Bp[q * 16 + Nn];
            }
        }
        // ---- compute with current fragments ----
#pragma unroll
        for (int q = 0; q < NT; ++q)
            acc[q] = __builtin_amdgcn_wmma_f32_16x16x4_f32(
                         false, av, false, bv[q], (short)0, acc[q], false, false);
        // ---- rotate buffers ----
        av = avn;
#pragma unroll
        for (int q = 0; q < NT; ++q) bv[q] = bvn[q];
    }

#pragma unroll
    for (int q = 0; q < NT; ++q) {
        int col = tn + q * 16 + r16;
        float bcol = bias ? bias[col] : 0.0f;
#pragma unroll
        for (int i2 = 0; i2 < 8; ++i2) {
            float v = acc[q][i2] + bcol;
            if (act == 1)      v = fmaxf(v, 0.0f);
            else if (act == 2) v = 1.0f / (1.0f + expf(-v));
            C[(size_t)(tm + i2 + (hi << 3)) * Nn + col] = v;
        }
    }
}

// =====================================================================
// h = LayerNorm(relu_out + h) * g + b        (one block per row, H=512)
// =====================================================================
__global__ __launch_bounds__(256) void ln_res_kernel(
    const float* __restrict__ r, float* __restrict__ h,
    const float* __restrict__ g, const float* __restrict__ bb)
{
    int row = blockIdx.x;
    const float* rp = r + (size_t)row * HH;
    float* hp = h + (size_t)row * HH;
    int t = threadIdx.x;

    float t0 = rp[t] + hp[t];
    float t1 = rp[t + 256] + hp[t + 256];

    __shared__ float red[256];
    red[t] = t0 + t1;
    __syncthreads();
    for (int s = 128; s > 0; s >>= 1) { if (t < s) red[t] += red[t + s]; __syncthreads(); }
    float mu = red[0] * (1.0f / HH);
    __syncthreads();

    float d0 = t0 - mu, d1 = t1 - mu;
    red[t] = d0 * d0 + d1 * d1;
    __syncthreads();
    for (int s = 128; s > 0; s >>= 1) { if (t < s) red[t] += red[t + s]; __syncthreads(); }
    float inv = rsqrtf(red[0] * (1.0f / HH) + 1e-5f);

    hp[t]       = d0 * inv * g[t]       + bb[t];
    hp[t + 256] = d1 * inv * g[t + 256] + bb[t + 256];
}

// masked mean over nodes:  hg[b, :] = sum_n h[b,n,:]*mask / max(sum mask, 1)
__global__ __launch_bounds__(256) void hgraph_kernel(
    const float* __restrict__ h, const float* __restrict__ mask,
    float* __restrict__ hg)
{
    int b = blockIdx.x;
    float msum = 0.f;
    for (int n = 0; n < NN; ++n) msum += mask[b * NN + n];
    float inv = 1.0f / fmaxf(msum, 1.0f);
    for (int c = threadIdx.x; c < HH; c += 256) {
        float s = 0.f;
        for (int n = 0; n < NN; ++n)
            s += h[((size_t)b * NN + n) * HH + c] * mask[b * NN + n];
        hg[b * HH + c] = s * inv;
    }
}

// z = mu + eps * exp(0.5*logvar)
__global__ void z_kernel(const float* __restrict__ mu, const float* __restrict__ lv,
                         const float* __restrict__ eps, float* __restrict__ z, int n)
{
    int i = blockIdx.x * 256 + threadIdx.x;
    if (i < n) z[i] = mu[i] + eps[i] * expf(0.5f * lv[i]);
}

// scatter upper-tri edge probs into symmetric adj_pred; zero diagonal
__global__ void scatter_adj_kernel(const float* __restrict__ e, float* __restrict__ ap)
{
    int b = blockIdx.y;
    int p = blockIdx.x * 256 + threadIdx.x;
    if (p < NE) {
        int i = 0, cum = 0;
        while (p >= cum + (NN - 1 - i)) { cum += NN - 1 - i; ++i; }
        int j = i + 1 + (p - cum);
        float v = e[(size_t)b * NE + p];
        ap[((size_t)b * NN + i) * NN + j] = v;
        ap[((size_t)b * NN + j) * NN + i] = v;
    } else if (p < NE + NN) {
        int d = p - NE;
        ap[((size_t)b * NN + d) * NN + d] = 0.f;
    }
}

// negcost[b,i,j] = -( -||ap_i - a_j||^2 + npred[b,i] ) masked by column count
__global__ __launch_bounds__(64) void cost_kernel(
    const float* __restrict__ ap, const float* __restrict__ adj,
    const float* __restrict__ npred, const float* __restrict__ mask,
    float* __restrict__ ncost)
{
    int b = blockIdx.y, i = blockIdx.x, j = threadIdx.x;
    float nact = 0.f;
    for (int n = 0; n < NN; ++n) nact += mask[b * NN + n];
    float d2 = 0.f;
    const float* api = ap  + ((size_t)b * NN + i) * NN;
    const float* aj  = adj + ((size_t)b * NN + j) * NN;   // adj symmetric: row j == col j
    for (int k = 0; k < NN; ++k) { float df = api[k] - aj[k]; d2 += df * df; }
    float c = -d2 + npred[b * NN + i];
    if (j >= (int)(nact + 0.5f)) c = 0.f;
    ncost[((size_t)b * NN + i) * NN + j] = -c;
}

// =====================================================================
// Jonker-Volgenant Hungarian (minimize), one 64-thread block per graph.
// =====================================================================
#define HINF 1e30f
__global__ __launch_bounds__(64) void hungarian_kernel(
    const float* __restrict__ ncost, int* __restrict__ perms)
{
    int b = blockIdx.x;
    int t = threadIdx.x;                      // 0..63

    __shared__ float a[NN * NN];
    __shared__ float u[NN + 1], v[NN + 1], minv[NN + 1];
    __shared__ int   p[NN + 1], way[NN + 1], used[NN + 1];
    __shared__ float rv[NN];
    __shared__ int   rj[NN];
    __shared__ int   s_j0;

    for (int k = t; k < NN * NN; k += 64) a[k] = ncost[(size_t)b * NN * NN + k];
    for (int k = t; k < NN + 1; k += 64) { u[k] = 0.f; v[k] = 0.f; p[k] = 0; way[k] = 0; }
    __syncthreads();

    for (int i = 1; i <= NN; ++i) {
        if (t == 0) { p[0] = i; s_j0 = 0; }
        for (int k = t; k < NN + 1; k += 64) { minv[k] = HINF; used[k] = 0; }
        __syncthreads();

        while (true) {
            if (t == 0) used[s_j0] = 1;
            __syncthreads();
            int j0 = s_j0;
            int i0 = p[j0];

            int j = t + 1;                    // columns 1..64
            float mv = HINF;
            if (!used[j]) {
                float cur = a[(i0 - 1) * NN + (j - 1)] - u[i0] - v[j];
                if (cur < minv[j]) { minv[j] = cur; way[j] = j0; }
                mv = minv[j];
            }
            rv[t] = mv; rj[t] = j;
            __syncthreads();
            for (int s = 32; s > 0; s >>= 1) {
                if (t < s && rv[t + s] < rv[t]) { rv[t] = rv[t + s]; rj[t] = rj[t + s]; }
                __syncthreads();
            }
            float delta = rv[0];
            int j1 = rj[0];
            __syncthreads();

            for (int k = t; k < NN + 1; k += 64) {
                if (used[k]) { u[p[k]] += delta; v[k] -= delta; }  // p[k] distinct over used k
                else minv[k] -= delta;
            }
            if (t == 0) s_j0 = j1;
            __syncthreads();
            if (p[j1] == 0) break;            // uniform across block
        }
        __syncthreads();
        if (t == 0) {                          // augment (serial)
            int j0 = s_j0;
            while (j0) { int j1 = way[j0]; p[j0] = p[j1]; j0 = j1; }
        }
        __syncthreads();
    }
    int j = t + 1;
    perms[(size_t)b * NN + (p[j] - 1)] = j - 1;   // row2col
}

// per-batch reconstruction BCE (edges on upper-tri logits + nodes)
__global__ __launch_bounds__(256) void bce_kernel(
    const float* __restrict__ e, const float* __restrict__ npred,
    const float* __restrict__ adj, const float* __restrict__ mask,
    const int* __restrict__ perms, float* __restrict__ out_pb)
{
    int b = blockIdx.x, t = threadIdx.x;
    __shared__ int pr[NN];
    __shared__ float red[256];
    if (t < NN) pr[t] = perms[b * NN + t];
    __syncthreads();

    float se = 0.f;
    for (int pp = t; pp < NE; pp += 256) {
        int i = 0, cum = 0;
        while (pp >= cum + (NN - 1 - i)) { cum += NN - 1 - i; ++i; }
        int j = i + 1 + (pp - cum);
        float pred = e[(size_t)b * NE + pp];
        pred = fminf(fmaxf(pred, 1e-6f), 1.0f - 1e-6f);
        float tg = adj[((size_t)b * NN + pr[i]) * NN + pr[j]];
        se -= tg * logf(pred) + (1.0f - tg) * logf(1.0f - pred);
    }
    float sn = 0.f;
    if (t < NN) {
        float pred = npred[b * NN + t];
        pred = fminf(fmaxf(pred, 1e-6f), 1.0f - 1e-6f);
        float tg = mask[b * NN + pr[t]];
        sn = -(tg * logf(pred) + (1.0f - tg) * logf(1.0f - pred));
    }
    red[t] = se * (1.0f / NE) + sn * (1.0f / NN);
    __syncthreads();
    for (int s = 128; s > 0; s >>= 1) { if (t < s) red[t] += red[t + s]; __syncthreads(); }
    if (t == 0) out_pb[b] = red[0];
}

// final: KL over mu/logvar + mean of per-batch recon -> [total, recon, kl]
__global__ __launch_bounds__(256) void final_kernel(
    const float* __restrict__ mu, const float* __restrict__ lv,
    const float* __restrict__ bcepb, float* __restrict__ out)
{
    __shared__ float red[256];
    __shared__ float kl_s;
    int t = threadIdx.x;

    float s = 0.f;
    for (int k = t; k < BN * LL; k += 256) {
        float l = lv[k], m = mu[k];
        s += 1.0f + l - m * m - expf(l);
    }
    red[t] = s;
    __syncthreads();
    for (int st = 128; st > 0; st >>= 1) { if (t < st) red[t] += red[t + st]; __syncthreads(); }
    if (t == 0) kl_s = -0.5f * red[0] / (float)(BN * LL);
    __syncthreads();

    red[t] = (t < BN) ? bcepb[t] : 0.f;
    __syncthreads();
    for (int st = 128; st > 0; st >>= 1) { if (t < st) red[t] += red[t + st]; __syncthreads(); }
    if (t == 0) {
        float recon = red[0] / (float)BN;
        out[0] = recon + kl_s;
        out[1] = recon;
        out[2] = kl_s;
    }
}

// =====================================================================
static inline void launch_gemm(const float* A, const float* Bm, const float* bias,
                               float* C, int M, int Nn, int K, int batch,
                               int sA, int sB, int sC, int act, hipStream_t stream)
{
    int tilesN = Nn / 16;
    int tilesM = M / 16;
    if ((tilesN & 3) == 0) {
        int strips = tilesM * (tilesN / 4);
        dim3 grid((strips + 7) / 8, batch);
        wmma_gemm_kernel<4><<<grid, 256, 0, stream>>>(A, Bm, bias, C, M, Nn, K, sA, sB, sC, act);
    } else if ((tilesN & 1) == 0) {
        int strips = tilesM * (tilesN / 2);
        dim3 grid((strips + 7) / 8, batch);
        wmma_gemm_kernel<2><<<grid, 256, 0, stream>>>(A, Bm, bias, C, M, Nn, K, sA, sB, sC, act);
    } else {
        int strips = tilesM * tilesN;
        dim3 grid((strips + 7) / 8, batch);
        wmma_gemm_kernel<1><<<grid, 256, 0, stream>>>(A, Bm, bias, C, M, Nn, K, sA, sB, sC, act);
    }
}

extern "C" void kernel_launch(void* const* d_in, const int* in_sizes, int n_in,
                              void* d_out, int out_size, void* d_ws, size_t ws_size,
                              hipStream_t stream)
{
    (void)in_sizes; (void)n_in; (void)out_size; (void)ws_size;

    const float* adj  = (const float*)d_in[0];
    const float* x    = (const float*)d_in[1];
    const float* mask = (const float*)d_in[2];
    const float* eps  = (const float*)d_in[3];
    const float* Wp   = (const float*)d_in[4];
    const float* bp   = (const float*)d_in[5];
    const float* Wc   = (const float*)d_in[6];
    const float* bc   = (const float*)d_in[7];
    const float* ln_g = (const float*)d_in[8];
    const float* ln_b = (const float*)d_in[9];
    const float* Wmu  = (const float*)d_in[10];
    const float* bmu  = (const float*)d_in[11];
    const float* Wlv  = (const float*)d_in[12];
    const float* blv  = (const float*)d_in[13];
    const float* Wd1  = (const float*)d_in[14];
    const float* bd1  = (const float*)d_in[15];
    const float* Wd2  = (const float*)d_in[16];
    const float* bd2  = (const float*)d_in[17];
    const float* Wadj = (const float*)d_in[18];
    const float* badj = (const float*)d_in[19];
    const float* Wnode= (const float*)d_in[20];
    const float* bnode= (const float*)d_in[21];

    float* f = (float*)d_ws;
    float* h     = f;                       // ROWS*HH   = 2097152
    float* m     = h     + (size_t)ROWS * HH;
    float* r     = m     + (size_t)ROWS * HH;
    float* hg    = r     + (size_t)ROWS * HH;     // BN*HH
    float* mu    = hg    + (size_t)BN * HH;       // BN*LL
    float* lv    = mu    + (size_t)BN * LL;
    float* z     = lv    + (size_t)BN * LL;
    float* hb1   = z     + (size_t)BN * LL;       // BN*HH
    float* hb2   = hb1   + (size_t)BN * HH;
    float* e     = hb2   + (size_t)BN * HH;       // BN*NE
    float* npred = e     + (size_t)BN * NE;       // BN*NN
    float* apred = npred + (size_t)BN * NN;       // BN*NN*NN
    float* ncost = apred + (size_t)BN * NN * NN;  // BN*NN*NN
    float* bcepb = ncost + (size_t)BN * NN * NN;  // BN
    int*   perms = (int*)(bcepb + BN);            // BN*NN

    // ---- encoder ----
    launch_gemm(x, Wp, bp, h, ROWS, HH, FD, 1, 0, 0, 0, 0, stream);   // h = x@Wp+bp
    for (int l = 0; l < 3; ++l) {
        // m[b] = adj[b] @ h[b]   (batched 64x512x64)
        launch_gemm(adj, h, nullptr, m, NN, HH, NN, BN,
                    NN * NN, NN * HH, NN * HH, 0, stream);
        // r = relu(m @ Wc[l] + bc[l])
        launch_gemm(m, Wc + (size_t)l * HH * HH, bc + l * HH, r,
                    ROWS, HH, HH, 1, 0, 0, 0, 1, stream);
        // h = LN(r + h)
        ln_res_kernel<<<ROWS, 256, 0, stream>>>(r, h, ln_g + l * HH, ln_b + l * HH);
    }
    hgraph_kernel<<<BN, 256, 0, stream>>>(h, mask, hg);

    launch_gemm(hg, Wmu, bmu, mu, BN, LL, HH, 1, 0, 0, 0, 0, stream);
    launch_gemm(hg, Wlv, blv, lv, BN, LL, HH, 1, 0, 0, 0, 0, stream);
    z_kernel<<<(BN * LL + 255) / 256, 256, 0, stream>>>(mu, lv, eps, z, BN * LL);

    // ---- decoder ----
    launch_gemm(z,   Wd1,  bd1,  hb1,  BN, HH, LL, 1, 0, 0, 0, 1, stream);
    launch_gemm(hb1, Wd2,  bd2,  hb2,  BN, HH, HH, 1, 0, 0, 0, 1, stream);
    launch_gemm(hb2, Wadj, badj, e,    BN, NE, HH, 1, 0, 0, 0, 2, stream);   // NT=2 path
    launch_gemm(hb2, Wnode,bnode,npred,BN, NN, HH, 1, 0, 0, 0, 2, stream);

    // ---- matching ----
    {
        dim3 g((NE + NN + 255) / 256, BN);
        scatter_adj_kernel<<<g, 256, 0, stream>>>(e, apred);
    }
    {
        dim3 g(NN, BN);
        cost_kernel<<<g, 64, 0, stream>>>(apred, adj, npred, mask, ncost);
    }
    hungarian_kernel<<<BN, 64, 0, stream>>>(ncost, perms);

    // ---- loss ----
    bce_kernel<<<BN, 256, 0, stream>>>(e, npred, adj, mask, perms, bcepb);
    final_kernel<<<1, 256, 0, stream>>>(mu, lv, bcepb, (float*)d_out);
}